// CenterLoss_83090437308894
// MI455X (gfx1250) — compile-verified
//
#include <hip/hip_runtime.h>

typedef __attribute__((ext_vector_type(2))) float v2f;
typedef __attribute__((ext_vector_type(8))) float v8f;

#define ALPHA  0.5f
#define BETA   0.1f
#define MARGIN 50.0f
#define SCALE  1.0f
#define BB 512
#define DD 512
#define CC 100000

// ---------------- kernel 1: bulk copy centers -> new_centers (d_out+1) ----------------
// d_out+1 is only 4-byte aligned, so scalar b32 copies (coalesced: 128B/wave32).
__global__ void copy_centers_k(const float* __restrict__ centers, float* __restrict__ outC) {
    size_t idx = (size_t)blockIdx.x * blockDim.x + threadIdx.x;
    size_t n = (size_t)CC * DD;
    if (idx < n) outC[idx] = centers[idx];
}

// ---------------- kernel 2: gather cg = centers[labels], fused loss partials ----------
__global__ void gather_loss_k(const float* __restrict__ features, const int* __restrict__ labels,
                              const float* __restrict__ centers, float* __restrict__ cg,
                              float* __restrict__ partial) {
    __shared__ float red[128];
    int i = blockIdx.x;
    int t = threadIdx.x;                        // 128 threads, 4 floats each
    int lab = labels[i];
    const float4* src = (const float4*)(centers + (size_t)lab * DD);
    const float4* f   = (const float4*)(features + (size_t)i * DD);
    float4*       dst = (float4*)(cg + (size_t)i * DD);
    float4 c  = src[t];
    float4 fv = f[t];
    dst[t] = c;
    float s = 0.f, dx;
    dx = c.x - fv.x; s += fminf(fmaxf(dx * dx, 1e-12f), 1e12f);
    dx = c.y - fv.y; s += fminf(fmaxf(dx * dx, 1e-12f), 1e12f);
    dx = c.z - fv.z; s += fminf(fmaxf(dx * dx, 1e-12f), 1e12f);
    dx = c.w - fv.w; s += fminf(fmaxf(dx * dx, 1e-12f), 1e12f);
    red[t] = s;
    __syncthreads();
    for (int off = 64; off > 0; off >>= 1) {
        if (t < off) red[t] += red[t + off];
        __syncthreads();
    }
    if (t == 0) partial[i] = red[0];
}

// ---------------- kernel 3: last-occurrence flags (JAX scatter: last write wins) ------
__global__ void flags_k(const int* __restrict__ labels, int* __restrict__ flags,
                        int* __restrict__ maskCnt) {
    int i = blockIdx.x * blockDim.x + threadIdx.x;
    if (i == 0) *maskCnt = 0;
    if (i < BB) {
        int lab = labels[i];
        int fl = 1;
        for (int j = i + 1; j < BB; ++j)
            if (labels[j] == lab) { fl = 0; break; }
        flags[i] = fl;
    }
}

// ---------------- kernel 4: deterministic loss tree-reduce -> d_out[0] ----------------
__global__ void loss_reduce_k(const float* __restrict__ partial, float* __restrict__ out0) {
    __shared__ float red[256];
    int t = threadIdx.x;
    red[t] = partial[t] + partial[t + 256];
    __syncthreads();
    for (int off = 128; off > 0; off >>= 1) {
        if (t < off) red[t] += red[t + off];
        __syncthreads();
    }
    if (t == 0) out0[0] = red[0] * (SCALE / (float)(BB * DD));
}

// ---------------- kernel 5: G = cg @ cg^T via V_WMMA_F32_16X16X4_F32 ------------------
// One wave per 16x16 output tile; 4 waves/block; grid (8, 32) covers 32x32 tiles.
__global__ void gram_wmma_k(const float* __restrict__ cg, float* __restrict__ G) {
    int wid  = threadIdx.x >> 5;
    int lane = threadIdx.x & 31;
    int tm   = blockIdx.y;
    int tn   = blockIdx.x * 4 + wid;
    int half = lane >> 4;   // selects K pair (ISA A/B f32 layout)
    int l16  = lane & 15;   // M row for A, N col for B
    const float* rowA = cg + (size_t)(tm * 16 + l16) * DD;
    const float* rowB = cg + (size_t)(tn * 16 + l16) * DD;  // B[k,n] = cg[n][k] (symmetric)
    v8f acc = {};
    for (int k0 = 0; k0 < DD; k0 += 4) {
        int k = k0 + half * 2;
        v2f a; a.x = rowA[k]; a.y = rowA[k + 1];
        v2f b; b.x = rowB[k]; b.y = rowB[k + 1];
        acc = __builtin_amdgcn_wmma_f32_16x16x4_f32(false, a, false, b, (short)0, acc,
                                                    false, false);
    }
    int rowbase = tm * 16 + half * 8;
    int col     = tn * 16 + l16;
#pragma unroll
    for (int r = 0; r < 8; ++r)
        G[(size_t)(rowbase + r) * DD + col] = acc[r];
}

// ---------------- kernel 6: masked softmax weights per row ----------------------------
__global__ void weights_k(const float* __restrict__ G, const int* __restrict__ labels,
                          float* __restrict__ W, float* __restrict__ rowsumW,
                          int* __restrict__ maskCnt) {
    __shared__ float red[256];
    __shared__ int   redi[256];
    __shared__ float sminv, sZ;
    int i = blockIdx.x;
    int t = threadIdx.x;
    int li = labels[i];
    float Gii = G[(size_t)i * DD + i];
    float negd[2], mk[2];
#pragma unroll
    for (int u = 0; u < 2; ++u) {
        int j = t + u * 256;
        float Gjj  = G[(size_t)j * DD + j];
        float d2   = fmaxf(Gii + Gjj - 2.f * G[(size_t)i * DD + j], 0.f);
        float dist = sqrtf(d2);
        bool  m    = (labels[j] != li) && (dist <= MARGIN);
        mk[u]   = m ? 1.f : 0.f;
        negd[u] = -dist;
    }
    // min_v = min_j (negdist * mask)   (masked entries contribute 0, as in reference)
    red[t] = fminf(negd[0] * mk[0], negd[1] * mk[1]);
    __syncthreads();
    for (int off = 128; off > 0; off >>= 1) {
        if (t < off) red[t] = fminf(red[t], red[t + off]);
        __syncthreads();
    }
    if (t == 0) sminv = red[0];
    __syncthreads();
    float minv = sminv;
    float num0 = expf(negd[0] - minv) * mk[0];
    float num1 = expf(negd[1] - minv) * mk[1];
    __syncthreads();                       // protect red[] reuse
    red[t]  = num0 + num1;
    redi[t] = (int)mk[0] + (int)mk[1];
    __syncthreads();
    for (int off = 128; off > 0; off >>= 1) {
        if (t < off) { red[t] += red[t + off]; redi[t] += redi[t + off]; }
        __syncthreads();
    }
    if (t == 0) {
        float Z = red[0] + 1e-6f;
        sZ = Z;
        rowsumW[i] = red[0] / Z;
        atomicAdd(maskCnt, redi[0]);       // integer atomic: deterministic
    }
    __syncthreads();
    float Z = sZ;
    W[(size_t)i * DD + t]       = num0 / Z;
    W[(size_t)i * DD + t + 256] = num1 / Z;
}

// ---------------- kernel 7: WC = W @ cg (WMMA f32), fuse new_vals + scatter -----------
__global__ void final_wmma_k(const float* __restrict__ W, const float* __restrict__ cg,
                             const float* __restrict__ features, const int* __restrict__ labels,
                             const float* __restrict__ rowsumW, const int* __restrict__ flags,
                             const int* __restrict__ maskCnt, float* __restrict__ outC) {
    int wid  = threadIdx.x >> 5;
    int lane = threadIdx.x & 31;
    int tm   = blockIdx.y;
    int tn   = blockIdx.x * 4 + wid;
    int half = lane >> 4;
    int l16  = lane & 15;
    const float* rowA = W  + (size_t)(tm * 16 + l16) * DD;  // A[m,k] = W[row][k]
    const float* colB = cg + (tn * 16 + l16);               // B[k,n] = cg[k][col]
    v8f acc = {};
    for (int k0 = 0; k0 < DD; k0 += 4) {
        int k = k0 + half * 2;
        v2f a; a.x = rowA[k];                a.y = rowA[k + 1];
        v2f b; b.x = colB[(size_t)k * DD];   b.y = colB[(size_t)(k + 1) * DD];
        acc = __builtin_amdgcn_wmma_f32_16x16x4_f32(false, a, false, b, (short)0, acc,
                                                    false, false);
    }
    float dfac = (*maskCnt > 0) ? BETA : 0.f;   // delta2 zeroed if mask.sum() < 1
    int col = tn * 16 + l16;
#pragma unroll
    for (int r = 0; r < 8; ++r) {
        int i = tm * 16 + half * 8 + r;
        float cgv    = cg[(size_t)i * DD + col];
        float fv     = features[(size_t)i * DD + col];
        float delta2 = rowsumW[i] * cgv - acc[r];              // sum_j W_ij (cg_i - cg_j)
        float newv   = cgv - ALPHA * (cgv - fv) - dfac * delta2;
        if (flags[i])                                          // last-occurrence only
            outC[(size_t)labels[i] * DD + col] = newv;
    }
}

extern "C" void kernel_launch(void* const* d_in, const int* in_sizes, int n_in,
                              void* d_out, int out_size, void* d_ws, size_t ws_size,
                              hipStream_t stream) {
    const float* features = (const float*)d_in[0];
    const int*   labels   = (const int*)d_in[1];
    const float* centers  = (const float*)d_in[2];
    float* out  = (float*)d_out;   // out[0] = loss, out+1 = new_centers (flat)
    float* outC = out + 1;

    char*  ws      = (char*)d_ws;
    float* cg      = (float*)(ws);                           // 1 MB
    float* G       = (float*)(ws + (1u << 20));              // 1 MB
    float* Wm      = (float*)(ws + (2u << 20));              // 1 MB
    float* rowsumW = (float*)(ws + (3u << 20));              // 2 KB
    float* partial = (float*)(ws + (3u << 20) + 2048);       // 2 KB
    int*   maskCnt = (int*)  (ws + (3u << 20) + 4096);       // 4 B
    int*   flags   = (int*)  (ws + (3u << 20) + 8192);       // 2 KB

    size_t nC = (size_t)CC * DD;
    copy_centers_k<<<dim3((unsigned)((nC + 255) / 256)), dim3(256), 0, stream>>>(centers, outC);
    gather_loss_k<<<dim3(BB), dim3(128), 0, stream>>>(features, labels, centers, cg, partial);
    flags_k<<<dim3(2), dim3(256), 0, stream>>>(labels, flags, maskCnt);
    loss_reduce_k<<<dim3(1), dim3(256), 0, stream>>>(partial, out);
    gram_wmma_k<<<dim3(8, 32), dim3(128), 0, stream>>>(cg, G);
    weights_k<<<dim3(BB), dim3(256), 0, stream>>>(G, labels, Wm, rowsumW, maskCnt);
    final_wmma_k<<<dim3(8, 32), dim3(128), 0, stream>>>(Wm, cg, features, labels, rowsumW,
                                                        flags, maskCnt, outC);
}